// GraphTransformerModel_81286551044271
// MI455X (gfx1250) — compile-verified
//
#include <hip/hip_runtime.h>
#include <hip/hip_bf16.h>

typedef __attribute__((ext_vector_type(2))) float v2f;
typedef __attribute__((ext_vector_type(8))) float v8f;

#define HDIM 64
#define EVOCAB 16
#define ENC_NEG_INF 0x007FFFFFu   // enc_ordf(-inf)

// Order-preserving encode for float -> uint so integer atomicMax == float max.
__device__ __forceinline__ unsigned enc_ordf(float f) {
  unsigned u = __float_as_uint(f);
  return (u & 0x80000000u) ? ~u : (u | 0x80000000u);
}
__device__ __forceinline__ float dec_ordf(unsigned e) {
  unsigned u = (e & 0x80000000u) ? (e ^ 0x80000000u) : ~e;
  return __uint_as_float(u);
}

// ---------------------------------------------------------------------------
// Tiny 16x64x64 GEMM: tab[v][h] = sum_d edge_emb[v][d] * We[d][h]
// (algebraic elimination of the per-edge lin_edge GEMM: ee = tab[edge_attr])
// ---------------------------------------------------------------------------
__global__ __launch_bounds__(256)
void build_eetab(const float* __restrict__ emb, const float* __restrict__ We,
                 float* __restrict__ tab) {
  int idx = blockIdx.x * blockDim.x + threadIdx.x;   // 16*64 = 1024
  if (idx >= EVOCAB * HDIM) return;
  int v = idx >> 6, h = idx & 63;
  float a = 0.f;
  #pragma unroll 8
  for (int d = 0; d < HDIM; ++d) a += emb[v * HDIM + d] * We[d * HDIM + h];
  tab[idx] = a;
}

// ---------------------------------------------------------------------------
// Per-layer state init: Acc = 0, amax = enc(-inf), denom = 0
// ---------------------------------------------------------------------------
__global__ __launch_bounds__(256)
void init_layer(float* __restrict__ Acc, unsigned* __restrict__ amaxE,
                float* __restrict__ denom, int N) {
  int i = blockIdx.x * blockDim.x + threadIdx.x;
  if (i < N * HDIM) Acc[i] = 0.f;
  if (i < N) { amaxE[i] = ENC_NEG_INF; denom[i] = 0.f; }
}

// ---------------------------------------------------------------------------
// Fused Q/K/V/skip projection:  [N, din] x [din, 4*64] (+bias) via
// v_wmma_f32_16x16x4_f32.
// Grid: (N/16, 4); blockIdx.y selects which of {Q,K,V,S}; the block's 4 waves
// split the 64 output columns into 16-col tiles and SHARE one 16xdin A tile,
// staged once into LDS with global_load_async_to_lds_b128 (ASYNCcnt path).
// A frag (16x4 f32, 2 VGPR): lanes 0-15 hold K=0,1; lanes 16-31 hold K=2,3.
// B frag mirrors A: lanes 0-15 hold rows K=0,1 at col=lane; 16-31 rows K=2,3.
// C/D (8 VGPR): VGPR r -> M=r (lanes 0-15) / M=r+8 (lanes 16-31), N=lane&15.
// ---------------------------------------------------------------------------
__global__ __launch_bounds__(128)
void fused_qkvs_gemm(const float* __restrict__ X, int din,
                     const float* __restrict__ Wq, const float* __restrict__ bq,
                     const float* __restrict__ Wk, const float* __restrict__ bk,
                     const float* __restrict__ Wv, const float* __restrict__ bv,
                     const float* __restrict__ Ws, const float* __restrict__ bs,
                     float* __restrict__ Q, float* __restrict__ K,
                     float* __restrict__ V, float* __restrict__ S) {
  __shared__ float As[16 * 128];               // A tile (up to 16x128 f32, 8 KB)

  const int wave = threadIdx.x >> 5;
  const int lane = threadIdx.x & 31;
  const int l15  = lane & 15;
  const int half = lane >> 4;                  // 0: lanes 0-15, 1: lanes 16-31
  const int wsel = blockIdx.y;                 // which of {Q,K,V,S}
  const int csub = wave << 4;                  // 16-col subtile within 64

  const float* W; const float* B; float* O;
  switch (wsel) {
    case 0:  W = Wq; B = bq; O = Q; break;
    case 1:  W = Wk; B = bk; O = K; break;
    case 2:  W = Wv; B = bv; O = V; break;
    default: W = Ws; B = bs; O = S; break;
  }

  const int row0 = blockIdx.x << 4;

  // ---- async-stage the contiguous 16 x din A tile into LDS ----
  {
    const char* gbase = (const char*)(X + (size_t)row0 * din);
    const unsigned lbase = (unsigned)(uintptr_t)(void*)As;  // low 32b = LDS offset
    const int nchunk = din * 4;                // 16-byte chunks in the tile
    for (int c = threadIdx.x; c < nchunk; c += 128) {
      unsigned loff = lbase + (unsigned)c * 16u;
      const void* ga = gbase + (size_t)c * 16u;
      asm volatile("global_load_async_to_lds_b128 %0, %1, off"
                   :: "v"(loff), "v"(ga) : "memory");
    }
    asm volatile("s_wait_asynccnt 0x0" ::: "memory");
    __syncthreads();
  }

  v8f acc = {0.f, 0.f, 0.f, 0.f, 0.f, 0.f, 0.f, 0.f};

  for (int kk = 0; kk < din; kk += 4) {
    const float* arow = As + l15 * din + kk + half * 2;     // ds_load_b64
    v2f a; a.x = arow[0]; a.y = arow[1];
    const float* bcol = W + (size_t)(kk + half * 2) * HDIM + csub + l15;
    v2f b; b.x = bcol[0]; b.y = bcol[HDIM];
    acc = __builtin_amdgcn_wmma_f32_16x16x4_f32(
        /*neg_a=*/false, a, /*neg_b=*/false, b,
        /*c_mod=*/(short)0, acc, /*reuse_a=*/false, /*reuse_b=*/false);
  }

  const float bias  = B[csub + l15];
  const int   rbase = row0 + half * 8;
  #pragma unroll
  for (int r = 0; r < 8; ++r)
    O[(size_t)(rbase + r) * HDIM + csub + l15] = acc[r] + bias;
}

// ---------------------------------------------------------------------------
// Edge pass 1: alpha_e = <q[dst], k[src] + eetab[attr]> / sqrt(64);
//              segment-max via encoded uint atomicMax. One wave per edge.
// ---------------------------------------------------------------------------
__global__ __launch_bounds__(256)
void edge_alpha_max(const int* __restrict__ src, const int* __restrict__ dst,
                    const int* __restrict__ eattr,
                    const float* __restrict__ Q, const float* __restrict__ K,
                    const float* __restrict__ tab,
                    float* __restrict__ alpha, unsigned* __restrict__ amaxE,
                    int E) {
  const int lane = threadIdx.x & 31;
  const int e = blockIdx.x * (blockDim.x >> 5) + (threadIdx.x >> 5);
  if (e >= E) return;
  // indices are wave-uniform: force SALU addressing (SGPR-base loads)
  const int s = __builtin_amdgcn_readfirstlane(src[e]);
  const int d = __builtin_amdgcn_readfirstlane(dst[e]);
  const int t = __builtin_amdgcn_readfirstlane(eattr[e]);
  const float2 q  = ((const float2*)(Q   + (size_t)d * HDIM))[lane];
  const float2 k  = ((const float2*)(K   + (size_t)s * HDIM))[lane];
  const float2 ee = ((const float2*)(tab + (size_t)t * HDIM))[lane];
  float p = q.x * (k.x + ee.x) + q.y * (k.y + ee.y);
  #pragma unroll
  for (int off = 16; off > 0; off >>= 1) p += __shfl_xor(p, off, 32);
  if (lane == 0) {
    p *= 0.125f;                                   // 1/sqrt(64)
    alpha[e] = p;
    atomicMax(amaxE + d, enc_ordf(p));
  }
}

// ---------------------------------------------------------------------------
// Edge pass 2: ea = exp(alpha - amax[dst]); denom[dst] += ea. Thread per edge.
// ---------------------------------------------------------------------------
__global__ __launch_bounds__(256)
void edge_exp_sum(const int* __restrict__ dst, float* __restrict__ alpha,
                  const unsigned* __restrict__ amaxE, float* __restrict__ denom,
                  int E) {
  int e = blockIdx.x * blockDim.x + threadIdx.x;
  if (e >= E) return;
  const int d = dst[e];
  const float ea = expf(alpha[e] - dec_ordf(amaxE[d]));
  alpha[e] = ea;
  atomicAdd(denom + d, ea);
}

// ---------------------------------------------------------------------------
// Edge pass 3: Acc[dst] += (v[src] + eetab[attr]) * ea/(denom[dst]+eps).
// One wave per edge, 2 channels per lane.
// ---------------------------------------------------------------------------
__global__ __launch_bounds__(256)
void edge_scatter(const int* __restrict__ src, const int* __restrict__ dst,
                  const int* __restrict__ eattr,
                  const float* __restrict__ V, const float* __restrict__ tab,
                  const float* __restrict__ alpha, const float* __restrict__ denom,
                  float* __restrict__ Acc, int E) {
  const int lane = threadIdx.x & 31;
  const int e = blockIdx.x * (blockDim.x >> 5) + (threadIdx.x >> 5);
  if (e >= E) return;
  const int s = __builtin_amdgcn_readfirstlane(src[e]);
  const int d = __builtin_amdgcn_readfirstlane(dst[e]);
  const int t = __builtin_amdgcn_readfirstlane(eattr[e]);
  const float coef = alpha[e] / (denom[d] + 1e-16f);
  const float2 v  = ((const float2*)(V   + (size_t)s * HDIM))[lane];
  const float2 ee = ((const float2*)(tab + (size_t)t * HDIM))[lane];
  float* ap = Acc + (size_t)d * HDIM + lane * 2;
  atomicAdd(ap,     (v.x + ee.x) * coef);
  atomicAdd(ap + 1, (v.y + ee.y) * coef);
}

// ---------------------------------------------------------------------------
// Epilogue: H = relu(Acc + Skip)   (skip already carries its bias)
// ---------------------------------------------------------------------------
__global__ __launch_bounds__(256)
void relu_skip(const float* __restrict__ Acc, const float* __restrict__ Sk,
               float* __restrict__ H, int n64) {
  int i = blockIdx.x * blockDim.x + threadIdx.x;
  if (i >= n64) return;
  float t = Acc[i] + Sk[i];
  H[i] = t > 0.f ? t : 0.f;
}

// ---------------------------------------------------------------------------
// Output projection: [N,64] x [64,2] + bias. Thread per node.
// ---------------------------------------------------------------------------
__global__ __launch_bounds__(256)
void out_proj(const float* __restrict__ H, const float* __restrict__ Wout,
              const float* __restrict__ bout, float* __restrict__ out, int N) {
  int n = blockIdx.x * blockDim.x + threadIdx.x;
  if (n >= N) return;
  float a0 = bout[0], a1 = bout[1];
  const float* h = H + (size_t)n * HDIM;
  #pragma unroll 8
  for (int c = 0; c < HDIM; ++c) {
    float hv = h[c];
    a0 += hv * Wout[c * 2 + 0];
    a1 += hv * Wout[c * 2 + 1];
  }
  out[n * 2 + 0] = a0;
  out[n * 2 + 1] = a1;
}

// ---------------------------------------------------------------------------
extern "C" void kernel_launch(void* const* d_in, const int* in_sizes, int n_in,
                              void* d_out, int out_size, void* d_ws, size_t ws_size,
                              hipStream_t stream) {
  const float* x        = (const float*)d_in[0];
  const int*   eidx     = (const int*)d_in[1];
  const int*   eattr    = (const int*)d_in[2];
  const float* edge_emb = (const float*)d_in[3];
  // c1: 4..12, c2: 13..21 in dict order Wq,bq,Wk,bk,Wv,bv,We,Wskip,bskip
  const float *c1_Wq = (const float*)d_in[4],  *c1_bq = (const float*)d_in[5];
  const float *c1_Wk = (const float*)d_in[6],  *c1_bk = (const float*)d_in[7];
  const float *c1_Wv = (const float*)d_in[8],  *c1_bv = (const float*)d_in[9];
  const float *c1_We = (const float*)d_in[10];
  const float *c1_Ws = (const float*)d_in[11], *c1_bs = (const float*)d_in[12];
  const float *c2_Wq = (const float*)d_in[13], *c2_bq = (const float*)d_in[14];
  const float *c2_Wk = (const float*)d_in[15], *c2_bk = (const float*)d_in[16];
  const float *c2_Wv = (const float*)d_in[17], *c2_bv = (const float*)d_in[18];
  const float *c2_We = (const float*)d_in[19];
  const float *c2_Ws = (const float*)d_in[20], *c2_bs = (const float*)d_in[21];
  const float *Wout  = (const float*)d_in[22], *bout  = (const float*)d_in[23];

  const int din1 = in_sizes[4] / HDIM;   // 128
  const int N    = in_sizes[0] / din1;   // 50000 (divisible by 16)
  const int E    = in_sizes[2];          // 1600000
  const size_t NH = (size_t)N * HDIM;

  const int* src = eidx;
  const int* dst = eidx + E;

  // workspace layout
  float*    Q     = (float*)d_ws;
  float*    K     = Q + NH;
  float*    V     = K + NH;
  float*    Sk    = V + NH;
  float*    Hb    = Sk + NH;
  float*    Acc   = Hb + NH;
  float*    denom = Acc + NH;
  unsigned* amaxE = (unsigned*)(denom + N);
  float*    alpha = (float*)(amaxE + N);
  float*    tab1  = alpha + E;
  float*    tab2  = tab1 + EVOCAB * HDIM;

  const int tpb = 256;
  const dim3 gGemm(N / 16, 4);
  const int gEdgeWave = (E + 7) / 8;          // 8 waves per 256-thread block
  const int gEdgeThr  = (E + tpb - 1) / tpb;
  const int gN64      = (int)((NH + tpb - 1) / tpb);
  const int gNode     = (N + tpb - 1) / tpb;

  // precompute edge-type tables (lin_edge folded into a 16x64 lookup)
  build_eetab<<<4, tpb, 0, stream>>>(edge_emb, c1_We, tab1);
  build_eetab<<<4, tpb, 0, stream>>>(edge_emb, c2_We, tab2);

  // ---- layer 1 (din = 128, input = x) ----
  init_layer<<<gN64, tpb, 0, stream>>>(Acc, amaxE, denom, N);
  fused_qkvs_gemm<<<gGemm, 128, 0, stream>>>(x, din1,
      c1_Wq, c1_bq, c1_Wk, c1_bk, c1_Wv, c1_bv, c1_Ws, c1_bs, Q, K, V, Sk);
  edge_alpha_max<<<gEdgeWave, tpb, 0, stream>>>(src, dst, eattr, Q, K, tab1,
                                                alpha, amaxE, E);
  edge_exp_sum<<<gEdgeThr, tpb, 0, stream>>>(dst, alpha, amaxE, denom, E);
  edge_scatter<<<gEdgeWave, tpb, 0, stream>>>(src, dst, eattr, V, tab1,
                                              alpha, denom, Acc, E);
  relu_skip<<<gN64, tpb, 0, stream>>>(Acc, Sk, Hb, (int)NH);

  // ---- layer 2 (din = 64, input = Hb) ----
  init_layer<<<gN64, tpb, 0, stream>>>(Acc, amaxE, denom, N);
  fused_qkvs_gemm<<<gGemm, 128, 0, stream>>>(Hb, HDIM,
      c2_Wq, c2_bq, c2_Wk, c2_bk, c2_Wv, c2_bv, c2_Ws, c2_bs, Q, K, V, Sk);
  edge_alpha_max<<<gEdgeWave, tpb, 0, stream>>>(src, dst, eattr, Q, K, tab2,
                                                alpha, amaxE, E);
  edge_exp_sum<<<gEdgeThr, tpb, 0, stream>>>(dst, alpha, amaxE, denom, E);
  edge_scatter<<<gEdgeWave, tpb, 0, stream>>>(src, dst, eattr, V, tab2,
                                              alpha, denom, Acc, E);
  relu_skip<<<gN64, tpb, 0, stream>>>(Acc, Sk, Hb, (int)NH);

  // ---- output projection ----
  out_proj<<<gNode, tpb, 0, stream>>>(Hb, Wout, bout, (float*)d_out, N);
}